// LayoutMetricLayer_2293512536231
// MI455X (gfx1250) — compile-verified
//
#include <hip/hip_runtime.h>

#define BATCH 256
#define SEQ   32
#define GRID  64
#define NLAB  24

typedef __attribute__((ext_vector_type(2))) float v2f;
typedef __attribute__((ext_vector_type(8))) float v8f;

__device__ __forceinline__ int imin(int a, int b) { return a < b ? a : b; }

__global__ __launch_bounds__(256)
void layout_metric_kernel(const float* __restrict__ pred_left,
                          const float* __restrict__ pred_top,
                          const float* __restrict__ pred_width,
                          const float* __restrict__ pred_height,
                          const float* __restrict__ pred_label,
                          const int*  __restrict__ true_left,
                          const int*  __restrict__ true_top,
                          const int*  __restrict__ true_width,
                          const int*  __restrict__ true_height,
                          const int*  __restrict__ true_label,
                          const int*  __restrict__ true_length,
                          const int*  __restrict__ pred_length,
                          float* __restrict__ out)
{
    __shared__ int tb[5][SEQ];                 // true: left, top, width, height, label
    __shared__ int pb[5][SEQ];                 // pred (argmax-decoded)
    __shared__ unsigned covy_t[GRID], covx_t[GRID], covy_p[GRID], covx_p[GRID];
    __shared__ int   conf[NLAB * NLAB];        // conf[pred][true]
    __shared__ float rowsum[32], colsum[32];   // padded to 32

    const int b   = blockIdx.x;
    const int tid = threadIdx.x;

    // gfx1250 prefetch of this batch's logits (global_prefetch_b8)
    if (tid == 0) {
        __builtin_prefetch(pred_left  + (size_t)b * SEQ * 64, 0, 1);
        __builtin_prefetch(pred_label + (size_t)b * SEQ * 24, 0, 1);
    }

    // zero confusion matrix
    for (int i = tid; i < NLAB * NLAB; i += 256) conf[i] = 0;

    // ---- Stage 1: decode pred boxes (argmax, first-max-wins) + load true boxes ----
    if (tid < 160) {
        const int field = tid >> 5;        // 0..4
        const int pos   = tid & 31;
        const float* base;
        int n;
        switch (field) {
            case 0: base = pred_left;   n = 64; break;
            case 1: base = pred_top;    n = 64; break;
            case 2: base = pred_width;  n = 64; break;
            case 3: base = pred_height; n = 64; break;
            default: base = pred_label; n = 24; break;
        }
        const float* p = base + ((size_t)b * SEQ + pos) * n;
        float best = p[0];
        int bi = 0;
        for (int i = 1; i < n; ++i) {
            float v = p[i];
            if (v > best) { best = v; bi = i; }   // strict >: first max wins (jnp.argmax)
        }
        pb[field][pos] = bi;
    } else if (tid < 192) {
        const int pos = tid - 160;
        const size_t o = (size_t)b * SEQ + pos;
        tb[0][pos] = true_left[o];
        tb[1][pos] = true_top[o];
        tb[2][pos] = true_width[o];
        tb[3][pos] = true_height[o];
        tb[4][pos] = true_label[o];
    }
    __syncthreads();

    // ---- Stage 2: separable coverage bitmasks over the 32 boxes ----
    {
        const int grp = tid >> 6;   // 0: covy_t, 1: covx_t, 2: covy_p, 3: covx_p
        const int c   = tid & 63;
        unsigned m = 0;
        if (grp == 0) {
            const int len = true_length[b];
            for (int j = 0; j < SEQ; ++j) {
                const int l = tb[0][j], t = tb[1][j];
                const int rgt = imin(GRID - 1, l + tb[2][j]);
                const int bot = imin(GRID - 1, t + tb[3][j]);
                const bool valid = (j < len) && (t < bot) && (l < rgt);
                if (valid && t <= c && c <= bot) m |= (1u << j);
            }
            covy_t[c] = m;
        } else if (grp == 1) {
            for (int j = 0; j < SEQ; ++j) {
                const int l = tb[0][j];
                const int rgt = imin(GRID - 1, l + tb[2][j]);
                if (l <= c && c <= rgt) m |= (1u << j);
            }
            covx_t[c] = m;
        } else if (grp == 2) {
            const int len = pred_length[b];
            for (int j = 0; j < SEQ; ++j) {
                const int l = pb[0][j], t = pb[1][j];
                const int rgt = imin(GRID - 1, l + pb[2][j]);
                const int bot = imin(GRID - 1, t + pb[3][j]);
                const bool valid = (j < len) && (t < bot) && (l < rgt);
                if (valid && t <= c && c <= bot) m |= (1u << j);
            }
            covy_p[c] = m;
        } else {
            for (int j = 0; j < SEQ; ++j) {
                const int l = pb[0][j];
                const int rgt = imin(GRID - 1, l + pb[2][j]);
                if (l <= c && c <= rgt) m |= (1u << j);
            }
            covx_p[c] = m;
        }
    }
    __syncthreads();

    // ---- Stage 3: rasterize both maps; last-box-wins = MSB of mask; histogram in LDS ----
    for (int p = tid; p < GRID * GRID; p += 256) {
        const int y = p >> 6, x = p & 63;
        const unsigned mt = covy_t[y] & covx_t[x];
        const unsigned mp = covy_p[y] & covx_p[x];
        const int lt = mt ? tb[4][31 - __builtin_clz(mt)] : 0;  // DEFAULT_LABEL = 0
        const int lp = mp ? pb[4][31 - __builtin_clz(mp)] : 0;
        atomicAdd(&conf[lp * NLAB + lt], 1);
    }
    __syncthreads();

    // ---- Stage 4: row/col sums of 24x24 confusion via chained V_WMMA_F32_16X16X4_F32 ----
    // A = 16x4 f32 chunk of conf (ISA layout: lanes 0-15 M, VGPR0/1 hold K,K+1; hi lanes K+2,K+3),
    // B = all-ones (layout independent) => every column of D = rowsum of A over chained K.
    if (tid < 32) {
        const int  lane = tid;
        const int  mloc = lane & 15;
        const bool hi   = lane >= 16;
        v2f ones = {1.0f, 1.0f};
        for (int rb = 0; rb < 2; ++rb) {
            const int r0 = rb * 16;
            v8f accr = {};   // rowsum of conf
            v8f accc = {};   // rowsum of conf^T == colsum of conf
            for (int kc = 0; kc < 6; ++kc) {
                const int k0 = kc * 4 + (hi ? 2 : 0);
                const int r  = r0 + mloc;
                v2f ar, ac;
                ar.x = (r < NLAB) ? (float)conf[r * NLAB + k0]       : 0.0f;
                ar.y = (r < NLAB) ? (float)conf[r * NLAB + k0 + 1]   : 0.0f;
                ac.x = (r < NLAB) ? (float)conf[k0 * NLAB + r]       : 0.0f;
                ac.y = (r < NLAB) ? (float)conf[(k0 + 1) * NLAB + r] : 0.0f;
                accr = __builtin_amdgcn_wmma_f32_16x16x4_f32(
                           false, ar, false, ones, (short)0, accr, false, false);
                accc = __builtin_amdgcn_wmma_f32_16x16x4_f32(
                           false, ac, false, ones, (short)0, accc, false, false);
            }
            // D layout: VGPR v -> M=v (lanes 0-15), M=v+8 (lanes 16-31); all N identical.
            if (lane == 0) {
#pragma unroll
                for (int v = 0; v < 8; ++v) { rowsum[r0 + v] = accr[v]; colsum[r0 + v] = accc[v]; }
            }
            if (lane == 16) {
#pragma unroll
                for (int v = 0; v < 8; ++v) { rowsum[r0 + 8 + v] = accr[v]; colsum[r0 + 8 + v] = accc[v]; }
            }
        }
    }
    __syncthreads();

    // ---- Stage 5: acc & mIoU ----
    if (tid == 0) {
        float inter_sum = 0.0f, wsum = 0.0f, iousum = 0.0f;
        for (int k = 0; k < NLAB; ++k) {
            const float d  = (float)conf[k * NLAB + k];
            const float un = rowsum[k] + colsum[k] - d;   // sum(axis=1)+sum(axis=2)-diag
            inter_sum += d;
            if (un > 0.0f) { wsum += 1.0f; iousum += d / (un + 1e-9f); }
        }
        out[b]         = inter_sum / (float)(GRID * GRID);  // confusion.sum() == 4096
        out[BATCH + b] = iousum / wsum;
    }
}

extern "C" void kernel_launch(void* const* d_in, const int* in_sizes, int n_in,
                              void* d_out, int out_size, void* d_ws, size_t ws_size,
                              hipStream_t stream) {
    (void)in_sizes; (void)n_in; (void)d_ws; (void)ws_size; (void)out_size;
    const float* pred_left   = (const float*)d_in[0];
    const float* pred_top    = (const float*)d_in[1];
    const float* pred_width  = (const float*)d_in[2];
    const float* pred_height = (const float*)d_in[3];
    const float* pred_label  = (const float*)d_in[4];
    const int*   true_left   = (const int*)d_in[5];
    const int*   true_top    = (const int*)d_in[6];
    const int*   true_width  = (const int*)d_in[7];
    const int*   true_height = (const int*)d_in[8];
    const int*   true_label  = (const int*)d_in[9];
    const int*   true_length = (const int*)d_in[10];
    const int*   pred_length = (const int*)d_in[11];
    float* out = (float*)d_out;

    layout_metric_kernel<<<BATCH, 256, 0, stream>>>(
        pred_left, pred_top, pred_width, pred_height, pred_label,
        true_left, true_top, true_width, true_height, true_label,
        true_length, pred_length, out);
}